// rl_Loss_61143154426508
// MI455X (gfx1250) — compile-verified
//
#include <hip/hip_runtime.h>
#include <hip/hip_bf16.h>

// Problem constants from the reference: prob (B,S,V) f32, target (B,S) int, reward (B,) f32.
#define BB 32
#define SS 128
#define VV 32000
// B*S = 4096 gathered elements total.

typedef __attribute__((ext_vector_type(2))) float v2f;
typedef __attribute__((ext_vector_type(8))) float v8f;

// Single-block kernel: 256 threads = 8 wave32s.
// Wave w handles chunks c = w*8 .. w*8+7; each chunk = 64 contiguous (b,s) elements.
// Per chunk, lane L supplies elements c*64+L and c*64+32+L as the two f32 A-matrix
// VGPRs of V_WMMA_F32_16X16X4_F32, with B = ones, so the matrix pipe performs the
// 64-way accumulation per instruction into the 16x16 f32 accumulator.
__global__ __launch_bounds__(256) void rl_loss_gather_wmma_kernel(
    const float* __restrict__ prob,
    const int*   __restrict__ target,
    const float* __restrict__ reward,
    float*       __restrict__ out)
{
    const int tid  = (int)threadIdx.x;
    const int lane = tid & 31;
    const int wave = tid >> 5;

    // ---- Phase 1: issue ALL gathered loads up front (max MLP, latency-bound) ----
    float xa[8];
    float xb[8];
#pragma unroll
    for (int i = 0; i < 8; ++i) {
        const int c  = wave * 8 + i;       // chunk id, 0..63
        const int e0 = c * 64 + lane;      // flat (b*S + s) index
        const int e1 = e0 + 32;
        const int t0 = target[e0];
        const int t1 = target[e1];
        const float r0 = reward[e0 >> 7];  // b = e / S, S == 128
        const float r1 = reward[e1 >> 7];
        xa[i] = prob[(size_t)e0 * (size_t)VV + (size_t)t0] * r0;
        xb[i] = prob[(size_t)e1 * (size_t)VV + (size_t)t1] * r1;
    }

    // ---- Phase 2: WMMA add-tree. D = A x ones(4x16) + C accumulates 64 values/instr ----
    const v2f bones = {1.0f, 1.0f};
    v8f acc = {};  // zero-initialized f32 16x16 accumulator (8 VGPRs)
#pragma unroll
    for (int i = 0; i < 8; ++i) {
        v2f a;
        a.x = xa[i];
        a.y = xb[i];
        // 8 args: (neg_a, A, neg_b, B, c_mod, C, reuse_a, reuse_b)
        acc = __builtin_amdgcn_wmma_f32_16x16x4_f32(
            /*neg_a=*/false, a, /*neg_b=*/false, bones,
            /*c_mod=*/(short)0, acc, /*reuse_a=*/false, /*reuse_b=*/false);
    }

    // C/D layout: VGPR r, lanes 0-15 hold M=r, lanes 16-31 hold M=r+8 (all N identical
    // since B was all-ones => each cell of row M equals row-sum S_M).
    // Per-lane sum over the 8 VGPRs: lanes 0-15 -> S_0+..+S_7, lanes 16-31 -> S_8+..+S_15.
    float t = acc[0] + acc[1] + acc[2] + acc[3] + acc[4] + acc[5] + acc[6] + acc[7];

    // ---- Phase 3: deterministic cross-wave reduction via LDS ----
    __shared__ float partial[16];  // 2 per wave (one per half-wave), 8 waves
    if (lane == 0 || lane == 16) {
        partial[wave * 2 + (lane >> 4)] = t;
    }
    __syncthreads();
    if (tid == 0) {
        float s = 0.0f;
#pragma unroll
        for (int i = 0; i < 16; ++i) s += partial[i];
        out[0] = -s;
    }
}

extern "C" void kernel_launch(void* const* d_in, const int* in_sizes, int n_in,
                              void* d_out, int out_size, void* d_ws, size_t ws_size,
                              hipStream_t stream) {
    (void)in_sizes; (void)n_in; (void)out_size; (void)d_ws; (void)ws_size;
    const float* prob   = (const float*)d_in[0];
    const int*   target = (const int*)  d_in[1];
    const float* reward = (const float*)d_in[2];
    float*       out    = (float*)d_out;

    // Single 256-thread block (8 wave32s): all 4096 gathers in flight at once;
    // problem is latency-bound, not bandwidth-bound, so one WGP suffices and the
    // single-block reduction stays bit-deterministic across replays.
    rl_loss_gather_wmma_kernel<<<1, 256, 0, stream>>>(prob, target, reward, out);
}